// timeGap_1365799600731
// MI455X (gfx1250) — compile-verified
//
#include <hip/hip_runtime.h>

// ---------------------------------------------------------------------------
// Problem constants (from the reference)
// ---------------------------------------------------------------------------
#define NUM_RGAP   32
#define NUM_SGAP   32
#define NUM_PCOUNT 64
#define EMB        256
#define INSZ       128            // NUM_RGAP + NUM_SGAP + NUM_PCOUNT
#define NTOK       (64 * 2048)    // B * L

#define BLOCKS     512
#define THREADS    256
#define NWAVES     (BLOCKS * (THREADS / 32))     // 4096 wave32s
#define NTILES     (NTOK / 16)                   // 8192 token tiles
#define NBSTRIPS   (EMB / 16)                    // 16 N-strips
#define TPW        ((NTILES * NBSTRIPS) / NWAVES) // 32 tiles per wave

typedef float v2f __attribute__((ext_vector_type(2)));
typedef float v8f __attribute__((ext_vector_type(8)));

// ---------------------------------------------------------------------------
// One-hot writers: value computed, never gathered -> pure coalesced f4 stores.
// ---------------------------------------------------------------------------
__global__ void onehot_f4_kernel(const int* __restrict__ idx,
                                 float4* __restrict__ out,
                                 int cols4, int total4) {
  int i = blockIdx.x * blockDim.x + threadIdx.x;
  if (i >= total4) return;
  int token = i / cols4;
  int c0 = (i - token * cols4) * 4;
  int v = idx[token];
  float4 r;
  r.x = (c0 + 0 == v) ? 1.0f : 0.0f;
  r.y = (c0 + 1 == v) ? 1.0f : 0.0f;
  r.z = (c0 + 2 == v) ? 1.0f : 0.0f;
  r.w = (c0 + 3 == v) ? 1.0f : 0.0f;
  out[i] = r;
}

// Fold one hot index into this lane's A-fragment bitmasks.
// Lane covers K = 4t + koff (+0 -> mx bit t, +1 -> my bit t).
__device__ __forceinline__ void addhot(int c, int koff,
                                       unsigned& mx, unsigned& my) {
  unsigned bit = 1u << (c >> 2);
  bool inHalf = ((c & 2) == koff);
  mx |= (inHalf && ((c & 1) == 0)) ? bit : 0u;
  my |= (inHalf && ((c & 1) == 1)) ? bit : 0u;
}

// ---------------------------------------------------------------------------
// tg_emb via precision-exact f32 WMMA.
// Each wave owns one 16-wide N-strip: B (16x128 slice of W^T) is loaded ONCE
// into 64 resident VGPRs, then 32 token tiles are swept with a memory-free
// WMMA chain (A built from bitmasks in VALU, co-executing with the XDL pipe).
// ---------------------------------------------------------------------------
__global__ void __launch_bounds__(THREADS)
emb_wmma_kernel(const int* __restrict__ rgap,
                const int* __restrict__ sgap,
                const int* __restrict__ pcount,
                const float* __restrict__ W,
                float* __restrict__ out_emb) {
  const int lane  = threadIdx.x & 31;
  const int wave  = threadIdx.x >> 5;
  const int gwave = blockIdx.x * (THREADS / 32) + wave;

  const int m    = lane & 15;     // A row / B col / D col within tile
  const int hi   = lane >> 4;     // 0 = lanes 0-15, 1 = lanes 16-31
  const int koff = hi << 1;       // K offset for upper half-wave

  const int nb       = (gwave & (NBSTRIPS - 1)) * 16;  // this wave's N-strip
  const int tileBase = (gwave / NBSTRIPS) * TPW;       // first token tile

  // ---- resident B fragments: 32 x v2f = 64 VGPRs, one-time load ----------
  // B[k][n] = W[nb+n][k]; pairs (k0,k0+1) are contiguous in W -> b64 loads.
  v2f b[32];
  {
    const float* wb = W + (size_t)(nb + m) * INSZ + koff;
#pragma unroll
    for (int t = 0; t < 32; ++t) {
      v2f x;
      x.x = wb[4 * t];
      x.y = wb[4 * t + 1];
      b[t] = x;
    }
  }

  // ---- software-pipelined token indices -----------------------------------
  int tile = tileBase;
  int tok  = tile * 16 + m;
  int c1 = rgap[tok];
  int c2 = NUM_RGAP + sgap[tok];
  int c3 = NUM_RGAP + NUM_SGAP + pcount[tok];

  for (int it = 0; it < TPW; ++it) {
    // prefetch next tile's indices (uniform select keeps EXEC all-ones)
    int ptile = (it + 1 < TPW) ? (tile + 1) : tile;
    int ptok  = ptile * 16 + m;
    int n1 = rgap[ptok];
    int n2 = NUM_RGAP + sgap[ptok];
    int n3 = NUM_RGAP + NUM_SGAP + pcount[ptok];

    // this lane's one-hot A encoded as two 32-bit masks
    unsigned mx = 0, my = 0;
    addhot(c1, koff, mx, my);
    addhot(c2, koff, mx, my);
    addhot(c3, koff, mx, my);

    // memory-free WMMA chain over K=128; dual accumulators hide XDL latency
    v8f acc0 = {}, acc1 = {};
#pragma unroll
    for (int t = 0; t < 32; t += 2) {
      v2f a0;
      a0.x = (float)((mx >> t) & 1u);
      a0.y = (float)((my >> t) & 1u);
      acc0 = __builtin_amdgcn_wmma_f32_16x16x4_f32(
          false, a0, false, b[t], (short)0, acc0, false, false);
      v2f a1;
      a1.x = (float)((mx >> (t + 1)) & 1u);
      a1.y = (float)((my >> (t + 1)) & 1u);
      acc1 = __builtin_amdgcn_wmma_f32_16x16x4_f32(
          false, a1, false, b[t + 1], (short)0, acc1, false, false);
    }
    v8f acc = acc0 + acc1;   // exact: partials hold disjoint hot terms

    // D layout: VGPR p -> row p (lanes 0-15) / p+8 (lanes 16-31)
    float* o = out_emb + (size_t)(tile * 16 + hi * 8) * EMB + nb + m;
#pragma unroll
    for (int p = 0; p < 8; ++p) {
      o[(size_t)p * EMB] = acc[p];
    }

    c1 = n1; c2 = n2; c3 = n3;
    ++tile;
  }
}

// ---------------------------------------------------------------------------
// Launch
// ---------------------------------------------------------------------------
extern "C" void kernel_launch(void* const* d_in, const int* in_sizes, int n_in,
                              void* d_out, int out_size, void* d_ws, size_t ws_size,
                              hipStream_t stream) {
  (void)in_sizes; (void)n_in; (void)out_size; (void)d_ws; (void)ws_size;

  const int*   rgap   = (const int*)d_in[0];
  const int*   sgap   = (const int*)d_in[1];
  const int*   pcount = (const int*)d_in[2];
  const float* W      = (const float*)d_in[3];

  float* out     = (float*)d_out;
  float* rg_out  = out;
  float* sg_out  = rg_out + (size_t)NTOK * NUM_RGAP;
  float* pc_out  = sg_out + (size_t)NTOK * NUM_SGAP;
  float* emb_out = pc_out + (size_t)NTOK * NUM_PCOUNT;

  // one-hot regions (pure store-bandwidth)
  {
    int cols4 = NUM_RGAP / 4, total4 = NTOK * cols4;
    onehot_f4_kernel<<<(total4 + 255) / 256, 256, 0, stream>>>(
        rgap, (float4*)rg_out, cols4, total4);
  }
  {
    int cols4 = NUM_SGAP / 4, total4 = NTOK * cols4;
    onehot_f4_kernel<<<(total4 + 255) / 256, 256, 0, stream>>>(
        sgap, (float4*)sg_out, cols4, total4);
  }
  {
    int cols4 = NUM_PCOUNT / 4, total4 = NTOK * cols4;
    onehot_f4_kernel<<<(total4 + 255) / 256, 256, 0, stream>>>(
        pcount, (float4*)pc_out, cols4, total4);
  }

  // embedding via register-resident-B f32 WMMA
  emb_wmma_kernel<<<BLOCKS, THREADS, 0, stream>>>(
      rgap, sgap, pcount, W, emb_out);
}